// GNNEncoder_15710990369655
// MI455X (gfx1250) — compile-verified
//
#include <hip/hip_runtime.h>
#include <hip/hip_bf16.h>

// ---------------------------------------------------------------------------
// GraphSAGE 2-layer encoder for MI455X (gfx1250, wave32, WMMA).
//  layer: out = relu?( mean_agg(x)@Wl + b + x@Wr )
// Memory-bound workload -> keep exact fp32, use V_WMMA_F32_16X16X4_F32.
// Weights staged into LDS pair-interleaved so every WMMA B fragment is one
// aligned ds_load_b64; B fragments for a K-step are prefetched into register
// arrays so DS latency overlaps the WMMA burst instead of stalling it.
// ---------------------------------------------------------------------------

#define N_NODES 50000
#define N_EDGES 800000
#define D_IN    64
#define D_HID   128
#define D_OUT   64

typedef __attribute__((ext_vector_type(2))) float v2f;
typedef __attribute__((ext_vector_type(8))) float v8f;

// ---------------------------- utility kernels ------------------------------

__global__ void gnn_zero_kernel(float* __restrict__ p, int n) {
    int i = blockIdx.x * blockDim.x + threadIdx.x;
    if (i < n) p[i] = 0.0f;
}

__global__ void gnn_degree_kernel(const long long* __restrict__ ei,
                                  float* __restrict__ deg) {
    int e = blockIdx.x * blockDim.x + threadIdx.x;
    if (e < N_EDGES) {
        int t = (int)ei[N_EDGES + e];
        atomicAdd(&deg[t], 1.0f);
    }
}

// scatter-add feat[src] into agg[tgt]; D threads cooperate per edge
template <int D>
__global__ void gnn_scatter_kernel(const float* __restrict__ feat,
                                   const long long* __restrict__ ei,
                                   float* __restrict__ agg) {
    constexpr int EPB = 256 / D;                  // edges per 256-thread block
    const int e = blockIdx.x * EPB + (threadIdx.x / D);
    const int f = threadIdx.x & (D - 1);
    if (e < N_EDGES) {
        const int s = (int)ei[e];
        const int t = (int)ei[N_EDGES + e];
        atomicAdd(&agg[t * D + f], feat[s * D + f]);
    }
}

// ------------------------------ fused GEMM ---------------------------------
// out[i,:] = act( (agg[i,:]/max(deg[i],1)) @ Wl + bias + xin[i,:] @ Wr )
// K = input feature dim, NOUT = output feature dim (both multiples of 16).
// 256 threads = 8 waves per block; each wave owns one 16-row M tile and all
// NOUT/16 N tiles.
//
// LDS weight layout (per matrix): pair-interleaved v2f
//   sW2[(k/2)*NOUT + col] = { W[k][col], W[k+1][col] }   (k even)
// so a B fragment for K-pair ka is one aligned 8-byte ds load:
//   lanes 0-15 get rows {k0,k0+1}, lanes 16-31 rows {k0+2,k0+3}.
template <int K, int NOUT, bool RELU>
__global__ __launch_bounds__(256)
void gnn_sage_gemm_kernel(const float* __restrict__ xin,
                          const float* __restrict__ agg,
                          const float* __restrict__ deg,
                          const float* __restrict__ Wl,
                          const float* __restrict__ Wr,
                          const float* __restrict__ bias,
                          float* __restrict__ out,
                          int nTiles) {
    constexpr int NT    = NOUT / 16;     // N tiles per wave
    constexpr int WTOT  = K * NOUT;      // floats per weight matrix
    constexpr int NC2   = NOUT / 2;      // column pairs
    constexpr int UNITS = (K / 2) * NC2; // float4 staging units per matrix
    __shared__ float sW[2 * WTOT];       // [0..WTOT) = Wl, [WTOT..) = Wr

    // Cooperative staging with pair-interleave relayout.
    // unit u -> row-pair k2, column-pair c2; writes one 16B-aligned float4.
    for (int u = threadIdx.x; u < UNITS; u += 256) {
        const int k2 = u / NC2;
        const int c2 = u - k2 * NC2;
        const int gofs = (2 * k2) * NOUT + 2 * c2;
        const int sofs = (k2 * NOUT + 2 * c2) * 2;
        float2 a0 = *(const float2*)&Wl[gofs];
        float2 a1 = *(const float2*)&Wl[gofs + NOUT];
        *(float4*)&sW[sofs] = make_float4(a0.x, a1.x, a0.y, a1.y);
        float2 b0 = *(const float2*)&Wr[gofs];
        float2 b1 = *(const float2*)&Wr[gofs + NOUT];
        *(float4*)&sW[WTOT + sofs] = make_float4(b0.x, b1.x, b0.y, b1.y);
    }
    __syncthreads();

    const int wave = threadIdx.x >> 5;            // 0..7
    const int ln   = threadIdx.x & 31;            // lane in wave32
    const int m    = blockIdx.x * 8 + wave;       // M tile index
    if (m >= nTiles) return;

    const int lrow  = ln & 15;                    // M within tile (A/B col idx)
    const int khalf = ln >> 4;                    // selects K pair {0,1} vs {2,3}
    const int row   = m * 16 + lrow;              // node row for A fragments

    const float rdeg = 1.0f / fmaxf(deg[row], 1.0f);

    v8f acc[NT];
#pragma unroll
    for (int n = 0; n < NT; ++n) acc[n] = (v8f){};

    const float* aggRow = agg + (long)row * K;
    const float* xRow   = xin + (long)row * K;
    const v2f* sBl = (const v2f*)sW;              // pair-interleaved Wl
    const v2f* sBr = (const v2f*)(sW + WTOT);     // pair-interleaved Wr

#pragma unroll 2
    for (int k0 = 0; k0 < K; k0 += 4) {
        const int ka = k0 + 2 * khalf;            // this half-wave's K pair
        const int kp = (k0 >> 1) + khalf;         // pair index ka/2
        // A fragment (16x4 f32): lanes 0-15 -> K = k0,k0+1 ; lanes 16-31 -> +2
        v2f aMean, aSelf;
        aMean.x = aggRow[ka]     * rdeg;
        aMean.y = aggRow[ka + 1] * rdeg;
        aSelf.x = xRow[ka];
        aSelf.y = xRow[ka + 1];

        // Prefetch the whole K-step's B fragments into registers so the DS
        // loads issue back-to-back and drain with partial dscnt waits while
        // the WMMA burst below executes.
        v2f bl[NT], br[NT];
#pragma unroll
        for (int n = 0; n < NT; ++n) {
            const int idx = kp * NOUT + n * 16 + lrow;   // v2f units
            bl[n] = sBl[idx];                            // ds_load_b64
            br[n] = sBr[idx];                            // ds_load_b64
        }

#pragma unroll
        for (int n = 0; n < NT; ++n) {
            // D = A*B + C, chained into the same accumulator
            acc[n] = __builtin_amdgcn_wmma_f32_16x16x4_f32(
                false, aMean, false, bl[n], (short)0, acc[n], false, false);
            acc[n] = __builtin_amdgcn_wmma_f32_16x16x4_f32(
                false, aSelf, false, br[n], (short)0, acc[n], false, false);
        }
    }

    // C/D layout: VGPR r -> M = r (lanes 0-15) / r+8 (lanes 16-31), N = lane&15
#pragma unroll
    for (int n = 0; n < NT; ++n) {
        const int col = n * 16 + lrow;
        const float bv = bias[col];
#pragma unroll
        for (int r = 0; r < 8; ++r) {
            const int orow = m * 16 + r + 8 * khalf;
            float v = acc[n][r] + bv;
            if (RELU) v = fmaxf(v, 0.0f);
            out[(long)orow * NOUT + col] = v;
        }
    }
}

// ------------------------------- launcher ----------------------------------

extern "C" void kernel_launch(void* const* d_in, const int* in_sizes, int n_in,
                              void* d_out, int out_size, void* d_ws, size_t ws_size,
                              hipStream_t stream) {
    const float*     x   = (const float*)d_in[0];
    const long long* ei  = (const long long*)d_in[1];   // int64 [2, N_EDGES]
    const float*     W1l = (const float*)d_in[2];
    const float*     b1  = (const float*)d_in[3];
    const float*     W1r = (const float*)d_in[4];
    const float*     W2l = (const float*)d_in[5];
    const float*     b2  = (const float*)d_in[6];
    const float*     W2r = (const float*)d_in[7];
    float* out = (float*)d_out;

    // workspace layout (floats): deg | agg1 | agg2 | h
    float* ws   = (float*)d_ws;
    float* deg  = ws;
    float* agg1 = deg  + N_NODES;                 // N_NODES * D_IN
    float* agg2 = agg1 + (size_t)N_NODES * D_IN;  // N_NODES * D_HID
    float* h    = agg2 + (size_t)N_NODES * D_HID; // N_NODES * D_HID
    (void)ws_size; (void)n_in; (void)in_sizes; (void)out_size;

    // 1) zero deg+agg1+agg2 (one contiguous region; atomics accumulate on top)
    {
        const int nz = N_NODES + N_NODES * D_IN + N_NODES * D_HID;
        gnn_zero_kernel<<<(nz + 255) / 256, 256, 0, stream>>>(ws, nz);
    }
    // 2) degrees
    gnn_degree_kernel<<<(N_EDGES + 255) / 256, 256, 0, stream>>>(ei, deg);
    // 3) layer-1 scatter: agg1[tgt] += x[src]
    {
        constexpr int EPB = 256 / D_IN;           // 4 edges per block
        gnn_scatter_kernel<D_IN><<<(N_EDGES + EPB - 1) / EPB, 256, 0, stream>>>(x, ei, agg1);
    }
    // 4) h = relu(mean1 @ W1l + b1 + x @ W1r)
    {
        const int tiles  = N_NODES / 16;          // 3125
        const int blocks = (tiles + 7) / 8;       // 391
        gnn_sage_gemm_kernel<D_IN, D_HID, true><<<blocks, 256, 0, stream>>>(
            x, agg1, deg, W1l, W1r, b1, h, tiles);
    }
    // 5) layer-2 scatter: agg2[tgt] += h[src]
    {
        constexpr int EPB = 256 / D_HID;          // 2 edges per block
        gnn_scatter_kernel<D_HID><<<(N_EDGES + EPB - 1) / EPB, 256, 0, stream>>>(h, ei, agg2);
    }
    // 6) out = mean2 @ W2l + b2 + h @ W2r
    {
        const int tiles  = N_NODES / 16;
        const int blocks = (tiles + 7) / 8;
        gnn_sage_gemm_kernel<D_HID, D_OUT, false><<<blocks, 256, 0, stream>>>(
            h, agg2, deg, W2l, W2r, b2, out, tiles);
    }
}